// HybridODEFunc_83915071030202
// MI455X (gfx1250) — compile-verified
//
#include <hip/hip_runtime.h>
#include <math.h>

typedef __attribute__((ext_vector_type(2))) float v2f;
typedef __attribute__((ext_vector_type(8))) float v8f;

#define ROWS_PER_BLOCK 256
#define HS 132   // padded LDS row stride in floats (528B: 16B aligned, low bank conflict)

__device__ __forceinline__ float fast_tanh(float x) {
#if __has_builtin(__builtin_amdgcn_tanhf)
  return __builtin_amdgcn_tanhf(x);
#elif __has_builtin(__builtin_amdgcn_tanh_f32)
  return __builtin_amdgcn_tanh_f32(x);
#else
  return tanhf(x);
#endif
}

__global__ __launch_bounds__(256)
void heart_ode_wmma_kernel(const float* __restrict__ tv,
                           const float* __restrict__ u,
                           const float* __restrict__ p,
                           const float* __restrict__ W1,
                           const float* __restrict__ b1,
                           const float* __restrict__ W2,
                           const float* __restrict__ b2,
                           float* __restrict__ out,
                           int Bn) {
  extern __shared__ float smem[];
  float* Hs  = smem;                         // ROWS_PER_BLOCK * HS floats
  float* W2s = smem + ROWS_PER_BLOCK * HS;   // 256 floats (W2 is 128x2)
  float* ps  = W2s + 256;                    // 37 floats

  const int tid  = threadIdx.x;
  const int lane = tid & 31;
  const int wave = tid >> 5;
  const int blockBase = blockIdx.x * ROWS_PER_BLOCK;

  // ---- stage small weights into LDS ----
  W2s[tid] = W2[tid];                 // exactly 256 elements
  if (tid < 37) ps[tid] = p[tid];

  const int hl = (lane < 16) ? 0 : 1; // lane half (ISA A/B/C layout split)
  const int nl = lane & 15;           // sub-index within half

  // ---- hoist B-matrix operands (W1 slices + bias row): grid-invariant ----
  // B 4x16 f32 layout: VGPR i, lane L -> B[k = i + 2*half][n = L%16]
  // K is padded 5 -> 8; slot k=5 carries the bias row (paired with constant-1
  // feature in A), so the C accumulator can start from inline 0.
  v2f bj0[8], bj1[8];
#pragma unroll
  for (int j = 0; j < 8; ++j) {
    const int n  = 16 * j + nl;
    const int k0 = 2 * hl;                 // step 0 covers k=0..3
    v2f b0; b0.x = W1[k0 * 128 + n]; b0.y = W1[(k0 + 1) * 128 + n];
    bj0[j] = b0;
    v2f bx;                                 // step 1 covers k=4..7
    bx.x = hl ? 0.0f : W1[4 * 128 + n];     // k=4: real feature row
    bx.y = hl ? 0.0f : b1[n];               // k=5: bias row
    bj1[j] = bx;
  }

  // ---- phase 1: H = tanh(X*W1 + b1), two 16-row tiles per wave ----
#pragma unroll
  for (int tile = 0; tile < 2; ++tile) {
    const int lr0 = wave * 32 + tile * 16;       // local row base of tile
    int row = blockBase + lr0 + nl;              // A layout: m = lane%16
    if (row >= Bn) row = Bn - 1;                 // clamp: keep EXEC all-ones
    const float* ur = u + (size_t)row * 10;
    // A 16x4 f32 layout: VGPR i, lane L -> A[m = L%16][k = i + 2*half]
    v2f a0; a0.x = ur[4 + 2 * hl]; a0.y = ur[5 + 2 * hl];   // features 0..3
    v2f a1;                                                  // k=4: feature 4
    a1.x = hl ? 0.0f : ur[8];                                // k=5: constant 1
    a1.y = hl ? 0.0f : 1.0f;                                 // (bias multiplier)

#pragma unroll
    for (int j = 0; j < 8; ++j) {
      v8f c = {};  // inline-0 C operand on first WMMA
      c = __builtin_amdgcn_wmma_f32_16x16x4_f32(false, a0, false, bj0[j],
                                                (short)0, c, false, false);
      c = __builtin_amdgcn_wmma_f32_16x16x4_f32(false, a1, false, bj1[j],
                                                (short)0, c, false, false);
      const int n = 16 * j + nl;
      // C/D layout: VGPR i, lane L -> D[m = i + 8*half][n = L%16]
#pragma unroll
      for (int i = 0; i < 8; ++i) {
        Hs[(lr0 + i + 8 * hl) * HS + n] = fast_tanh(c[i]);
      }
    }
  }

  __syncthreads();

  // ---- phase 2: z = H*W2 + b2, then ODE math; one row per thread ----
  const int row = blockBase + tid;
  float z0 = b2[0], z1 = b2[1];
  const float* hrow = Hs + tid * HS;             // 528B stride -> 16B aligned
#pragma unroll 8
  for (int k = 0; k < 128; k += 4) {
    float4 h4 = *(const float4*)(hrow + k);
    float4 wA = *(const float4*)(W2s + 2 * k);       // W2[k][0..1], W2[k+1][0..1]
    float4 wB = *(const float4*)(W2s + 2 * k + 4);   // W2[k+2][0..1], W2[k+3][0..1]
    z0 += h4.x * wA.x + h4.y * wA.z + h4.z * wB.x + h4.w * wB.z;
    z1 += h4.x * wA.y + h4.y * wA.w + h4.z * wB.y + h4.w * wB.w;
  }

  if (row < Bn) {
    const float* ur = u + (size_t)row * 10;
    const float Qmt = ur[0], Qav = ur[1], Qtc = ur[2], Qpv = ur[3];
    const float Vlv = ur[4], Vao = ur[5], Vvc = ur[6], Vrv = ur[7];
    const float Vpa = ur[8], Vpu = ur[9];

    const float Elvf = ps[0],  Eao  = ps[1],  Evc  = ps[2],  Ervf = ps[3];
    const float Epa  = ps[4],  Epu  = ps[5];
    const float Rmt  = ps[6],  Rav  = ps[7],  Rsys = ps[8],  Rtc  = ps[9];
    const float Rpv  = ps[10], Rpul = ps[11];
    const float Lmt  = ps[12], Lav  = ps[13], Ltc  = ps[14], Lpv  = ps[15];
    const float Vdlvf = ps[16], Vdao = ps[17], Vdvc = ps[18], Vdrvf = ps[19];
    const float Vdpa  = ps[20], Vdpu = ps[21];
    const float P0lvf = ps[22], P0rvf = ps[23];
    const float lamlvf = ps[24], lamrvf = ps[25];
    const float V0lvf = ps[27], V0rvf = ps[28];
    const float Pth = ps[36];

    const float tt = tv[row];
    const float ph = fmodf(tt, 0.75f) - 0.375f;
    const float e  = expf(-80.0f * ph * ph);

    const float Pperi = z0, Vspt = z1;
    const float Vlvf = Vlv - Vspt;
    const float Vrvf = Vrv + Vspt;
    const float exp_lv = expf(fminf(lamlvf * (Vlvf - V0lvf), 88.0f));
    const float exp_rv = expf(fminf(lamrvf * (Vrvf - V0rvf), 88.0f));
    const float Plvf = e * Elvf * (Vlvf - Vdlvf) + (1.0f - e) * P0lvf * (exp_lv - 1.0f);
    const float Prvf = e * Ervf * (Vrvf - Vdrvf) + (1.0f - e) * P0rvf * (exp_rv - 1.0f);
    const float Plv = Plvf + Pperi;
    const float Prv = Prvf + Pperi;
    const float Pao = Eao * (Vao - Vdao);
    const float Pvc = Evc * (Vvc - Vdvc);
    const float Ppa = Epa * (Vpa - Vdpa) + Pth;
    const float Ppu = Epu * (Vpu - Vdpu) + Pth;
    const float Qsys = (Pao - Pvc) / Rsys;
    const float Qpul = (Ppa - Ppu) / Rpul;

    const float du0 = ((Ppu - Plv > 0.0f) || (Qmt > 0.0f))
                        ? (Ppu - Plv - Qmt * Rmt) / Lmt : 0.0f;
    const float du1 = ((Plv - Pao > 0.0f) || (Qav > 0.0f))
                        ? (Plv - Pao - Qav * Rav) / Lav : 0.0f;
    const float du2 = ((Pvc - Prv > 0.0f) || (Qtc > 0.0f))
                        ? (Pvc - Prv - Qtc * Rtc) / Ltc : 0.0f;
    const float du3 = ((Prv - Ppa > 0.0f) || (Qpv > 0.0f))
                        ? (Prv - Ppa - Qpv * Rpv) / Lpv : 0.0f;
    const float Qmt_c = fmaxf(Qmt, 0.0f);
    const float Qav_c = fmaxf(Qav, 0.0f);
    const float Qtc_c = fmaxf(Qtc, 0.0f);
    const float Qpv_c = fmaxf(Qpv, 0.0f);

    float* o = out + (size_t)row * 10;
    o[0] = du0;
    o[1] = du1;
    o[2] = du2;
    o[3] = du3;
    o[4] = Qmt_c - Qav_c;
    o[5] = Qav_c - Qsys;
    o[6] = Qsys - Qtc_c;
    o[7] = Qtc_c - Qpv_c;
    o[8] = Qpv_c - Qpul;
    o[9] = Qpul - Qmt_c;
  }
}

extern "C" void kernel_launch(void* const* d_in, const int* in_sizes, int n_in,
                              void* d_out, int out_size, void* d_ws, size_t ws_size,
                              hipStream_t stream) {
  const float* t  = (const float*)d_in[0];
  const float* u  = (const float*)d_in[1];
  const float* p  = (const float*)d_in[2];
  const float* W1 = (const float*)d_in[3];
  const float* b1 = (const float*)d_in[4];
  const float* W2 = (const float*)d_in[5];
  const float* b2 = (const float*)d_in[6];
  float* out = (float*)d_out;

  const int Bn = in_sizes[0];
  const int blocks = (Bn + ROWS_PER_BLOCK - 1) / ROWS_PER_BLOCK;
  const size_t shmem = (size_t)(ROWS_PER_BLOCK * HS + 256 + 40) * sizeof(float);

  // CDNA5 WGP allows up to 320KB LDS per workgroup; raise the dynamic cap.
  (void)hipFuncSetAttribute((const void*)heart_ode_wmma_kernel,
                            hipFuncAttributeMaxDynamicSharedMemorySize,
                            (int)shmem);

  heart_ode_wmma_kernel<<<blocks, 256, shmem, stream>>>(t, u, p, W1, b1, W2, b2,
                                                        out, Bn);
}